// STD_10488310137025
// MI455X (gfx1250) — compile-verified
//
#include <hip/hip_runtime.h>
#include <math.h>

// ---------------------------------------------------------------------------
// FISTA sparse regression for MI455X (gfx1250), fp32 WMMA 16x16x4.
// x (2048x16384, 128MB) is L2-resident (192MB L2); all GEMMs are skinny
// (M=25 -> padded 32 = two 16-row WMMA tiles). 4 independent accumulator
// chains per wave keep the XDL pipe full. Deterministic: no atomics.
// ---------------------------------------------------------------------------

typedef __attribute__((ext_vector_type(2))) float v2f;
typedef __attribute__((ext_vector_type(8))) float v8f;

#define D_FEAT 2048
#define M_SIZE 16384
#define LROWS  25

// workspace offsets (floats); total 1,805,040 floats = 7.3 MB
#define OFF_A      0u
#define OFF_B      65536u
#define OFF_GRAD   131072u
#define OFF_R      196608u
#define OFF_AX     720896u
#define OFF_GRADP  1245184u
#define OFF_V      1769472u
#define OFF_W      1785856u
#define OFF_U      1802240u
#define OFF_WPART  1804288u
#define OFF_DTD    1804352u
#define OFF_YTAIL  1804992u
#define OFF_SCAL   1805024u

#define SC_TAU  0
#define SC_LAM2 1
#define SC_A    2
#define SC_RAT  3
#define SC_EIG  4
#define SC_SIG  5
#define SC_DONE 7

__device__ __forceinline__ v8f wmma4(v2f a, v2f b, v8f c) {
    // D = A(16x4,f32) * B(4x16,f32) + C(16x16,f32)
    return __builtin_amdgcn_wmma_f32_16x16x4_f32(
        false, a, false, b, (short)0, c, false, false);
}

// ---------------------------------------------------------------------------
// fill
__global__ void k_fill(float* __restrict__ p, int n, float val) {
    int i = blockIdx.x * blockDim.x + threadIdx.x;
    if (i < n) p[i] = val;
}

// ---------------------------------------------------------------------------
// init: build Delta (25x25), DtD = Delta^T Delta, power-iterate DtD for
// lambda_max (== _EIG_DELTA), init scalars and y-tail. Single block.
__global__ void k_init(float* __restrict__ ws) {
    __shared__ float Dl[25][25];
    __shared__ float Dt[25][25];
    __shared__ float vv[25], pp[25];
    __shared__ float n2s;
    float* DtDg  = ws + OFF_DTD;
    float* ytail = ws + OFF_YTAIL;
    float* scal  = ws + OFF_SCAL;
    int t = threadIdx.x;
    for (int e = t; e < 625; e += blockDim.x) {
        int i = e / 25, j = e % 25, d = i - j;
        float val = 0.f;
        if (d == 0) val = 1.f; else if (d == 1) val = -2.f; else if (d == 2) val = 1.f;
        if (i == 0 && j == 0) val = 0.f;
        if (i == 1 && j == 0) val = -1.f;
        Dl[i][j] = val;
    }
    __syncthreads();
    for (int e = t; e < 625; e += blockDim.x) {
        int i = e / 25, j = e % 25;
        float s = 0.f;
        for (int k = 0; k < 25; ++k) s += Dl[k][i] * Dl[k][j];
        Dt[i][j] = s;
        DtDg[e]  = s;
    }
    __syncthreads();
    if (t < 25) vv[t] = (t & 1) ? -0.2f : 0.2f;   // overlap with oscillatory top mode
    __syncthreads();
    for (int it = 0; it < 500; ++it) {
        if (t < 25) {
            float s = 0.f;
            for (int k = 0; k < 25; ++k) s += Dt[t][k] * vv[k];
            pp[t] = s;
        }
        __syncthreads();
        if (t == 0) {
            float s = 0.f;
            for (int k = 0; k < 25; ++k) s += pp[k] * pp[k];
            n2s = sqrtf(s) + 1e-30f;
        }
        __syncthreads();
        if (t < 25) vv[t] = pp[t] / n2s;
        __syncthreads();
    }
    if (t == 0) {
        scal[SC_EIG]  = n2s;    // lambda_max(DtD) = sigma_max(Delta)^2
        scal[SC_A]    = 1.f;
        scal[SC_RAT]  = 0.f;
        scal[SC_DONE] = 0.f;
        scal[SC_TAU]  = 0.f;
        scal[SC_LAM2] = 0.1f;
        scal[SC_SIG]  = 0.f;
    }
    if (t < 24) ytail[t] = 1.f;
}

// ---------------------------------------------------------------------------
// power iteration: u = x @ v  (one block per row, fixed-order LDS reduce)
__global__ void k_xv(const float* __restrict__ x, const float* __restrict__ v,
                     float* __restrict__ u) {
    __shared__ float red[256];
    const float* row = x + (size_t)blockIdx.x * M_SIZE;
    float s = 0.f;
    for (int j = threadIdx.x; j < M_SIZE; j += 256) s += row[j] * v[j];
    red[threadIdx.x] = s;
    __syncthreads();
    for (int off = 128; off; off >>= 1) {
        if (threadIdx.x < off) red[threadIdx.x] += red[threadIdx.x + off];
        __syncthreads();
    }
    if (threadIdx.x == 0) u[blockIdx.x] = red[0];
}

// w = x^T @ u, plus per-block partial of ||w||^2 (deterministic)
__global__ void k_xtw(const float* __restrict__ x, const float* __restrict__ u,
                      float* __restrict__ w, float* __restrict__ wpart) {
    int j = blockIdx.x * 256 + threadIdx.x;
    float s = 0.f;
    for (int i = 0; i < D_FEAT; ++i) s += x[(size_t)i * M_SIZE + j] * u[i];
    w[j] = s;
    __shared__ float red[256];
    red[threadIdx.x] = s * s;
    __syncthreads();
    for (int off = 128; off; off >>= 1) {
        if (threadIdx.x < off) red[threadIdx.x] += red[threadIdx.x + off];
        __syncthreads();
    }
    if (threadIdx.x == 0) wpart[blockIdx.x] = red[0];
}

__global__ void k_vnorm(const float* __restrict__ w, const float* __restrict__ wpart,
                        float* __restrict__ v) {
    float s = 0.f;
    for (int p = 0; p < 64; ++p) s += wpart[p];
    int j = blockIdx.x * 256 + threadIdx.x;
    v[j] = w[j] / (sqrtf(s) + 1e-12f);
}

// sigma^2 = ||u||^2 ; tau = 1/(2(sigma^2 + 0.1*eig)); lam2 select
__global__ void k_tau(const float* __restrict__ u, float* __restrict__ scal) {
    __shared__ float red[256];
    float s = 0.f;
    for (int i = threadIdx.x; i < D_FEAT; i += 256) s += u[i] * u[i];
    red[threadIdx.x] = s;
    __syncthreads();
    for (int off = 128; off; off >>= 1) {
        if (threadIdx.x < off) red[threadIdx.x] += red[threadIdx.x + off];
        __syncthreads();
    }
    if (threadIdx.x == 0) {
        float sig = red[0];
        scal[SC_SIG] = sig;
        float tau = 1.f / (2.f * (sig + 0.1f * scal[SC_EIG]));
        scal[SC_TAU]  = tau;
        scal[SC_LAM2] = (tau * 0.1f > 0.1f) ? (0.1f / tau) : 0.1f;
    }
}

// ---------------------------------------------------------------------------
// T = A(32x2048) @ x(2048x16384) via v_wmma_f32_16x16x4_f32.
// Block = 4 waves, each wave owns 16 columns; 2 row-tiles (M 0-15 / 16-31)
// x 2 K-interleaved partial sums = 4 independent WMMA chains (hides the
// multi-pass XDL latency). mode 0: store resid R = mask^2*(T - part_y);
// mode 1: store raw T into AX.
__global__ void k_gemm_ax(const float* __restrict__ x, const float* __restrict__ A,
                          const float* __restrict__ py, const float* __restrict__ mask,
                          float* __restrict__ out, const float* __restrict__ scal,
                          int mode) {
    if (scal[SC_DONE] != 0.f) return;          // uniform early-out (state frozen)
    const int lane = threadIdx.x & 31;
    const int wave = threadIdx.x >> 5;
    const int half = lane >> 4;
    const int lm   = lane & 15;
    const int n0   = blockIdx.x * 64 + wave * 16;
    const float* Ar0 = A + (size_t)lm * D_FEAT;
    const float* Ar1 = A + (size_t)(16 + lm) * D_FEAT;
    v8f acc0a = {}, acc0b = {};
    v8f acc1a = {}, acc1b = {};
#pragma unroll 2
    for (int k = 0; k < D_FEAT; k += 8) {
        int k0 = k + 2 * half;          // K pair {k0,k0+1}
        int k1 = k0 + 4;                // K pair {k1,k1+1}
        v2f a0, a1, b;
        a0.x = Ar0[k0]; a0.y = Ar0[k0 + 1];
        a1.x = Ar1[k0]; a1.y = Ar1[k0 + 1];
        b.x  = x[(size_t)k0 * M_SIZE + n0 + lm];
        b.y  = x[(size_t)(k0 + 1) * M_SIZE + n0 + lm];
        acc0a = wmma4(a0, b, acc0a);
        acc1a = wmma4(a1, b, acc1a);
        a0.x = Ar0[k1]; a0.y = Ar0[k1 + 1];
        a1.x = Ar1[k1]; a1.y = Ar1[k1 + 1];
        b.x  = x[(size_t)k1 * M_SIZE + n0 + lm];
        b.y  = x[(size_t)(k1 + 1) * M_SIZE + n0 + lm];
        acc0b = wmma4(a0, b, acc0b);
        acc1b = wmma4(a1, b, acc1b);
    }
    v8f acc0 = acc0a + acc0b;
    v8f acc1 = acc1a + acc1b;
    const int col = n0 + lm;
    if (mode == 0) {
        for (int r = 0; r < 8; ++r) {
            int m0 = r + 8 * half;        // 0..15
            int m1 = 16 + m0;             // 16..31
            float q = mask[(size_t)m0 * M_SIZE + col];
            out[(size_t)m0 * M_SIZE + col] = q * q * (acc0[r] - py[(size_t)m0 * M_SIZE + col]);
            if (m1 < LROWS) {
                q = mask[(size_t)m1 * M_SIZE + col];
                out[(size_t)m1 * M_SIZE + col] = q * q * (acc1[r] - py[(size_t)m1 * M_SIZE + col]);
            } else {
                out[(size_t)m1 * M_SIZE + col] = 0.f;   // keep pad rows clean
            }
        }
    } else {
        for (int r = 0; r < 8; ++r) {
            int m0 = r + 8 * half;
            int m1 = 16 + m0;
            out[(size_t)m0 * M_SIZE + col] = acc0[r];
            out[(size_t)m1 * M_SIZE + col] = acc1[r];
        }
    }
}

// ---------------------------------------------------------------------------
// grad = R(32x16384) @ x^T -> (32x2048). Split-K: grid (32 col-blocks, 8
// K-chunks of 2048); 4 WMMA chains per wave; partials reduced
// deterministically by k_gsum.
__global__ void k_gemm_rxt(const float* __restrict__ x, const float* __restrict__ R,
                           float* __restrict__ gradp, const float* __restrict__ scal) {
    if (scal[SC_DONE] != 0.f) return;
    const int lane = threadIdx.x & 31;
    const int wave = threadIdx.x >> 5;
    const int half = lane >> 4;
    const int lm   = lane & 15;
    const int n0   = blockIdx.x * 64 + wave * 16;   // columns of grad = rows of x
    const int kb   = blockIdx.y * 2048;             // K chunk
    const float* Rr0 = R + (size_t)lm * M_SIZE;
    const float* Rr1 = R + (size_t)(16 + lm) * M_SIZE;
    const float* xr  = x + (size_t)(n0 + lm) * M_SIZE;  // lane's own x row (B^T)
    v8f acc0a = {}, acc0b = {};
    v8f acc1a = {}, acc1b = {};
#pragma unroll 2
    for (int k = kb; k < kb + 2048; k += 8) {
        int k0 = k + 2 * half;
        int k1 = k0 + 4;
        v2f a0, a1, b;
        a0.x = Rr0[k0]; a0.y = Rr0[k0 + 1];
        a1.x = Rr1[k0]; a1.y = Rr1[k0 + 1];
        b.x  = xr[k0];  b.y  = xr[k0 + 1];
        acc0a = wmma4(a0, b, acc0a);
        acc1a = wmma4(a1, b, acc1a);
        a0.x = Rr0[k1]; a0.y = Rr0[k1 + 1];
        a1.x = Rr1[k1]; a1.y = Rr1[k1 + 1];
        b.x  = xr[k1];  b.y  = xr[k1 + 1];
        acc0b = wmma4(a0, b, acc0b);
        acc1b = wmma4(a1, b, acc1b);
    }
    v8f acc0 = acc0a + acc0b;
    v8f acc1 = acc1a + acc1b;
    const int col = n0 + lm;
    float* g = gradp + (size_t)blockIdx.y * 65536;
    for (int r = 0; r < 8; ++r) {
        int m0 = r + 8 * half;
        int m1 = 16 + m0;
        g[(size_t)m0 * D_FEAT + col] = acc0[r];
        g[(size_t)m1 * D_FEAT + col] = acc1[r];
    }
}

__global__ void k_gsum(const float* __restrict__ gradp, float* __restrict__ grad,
                       const float* __restrict__ scal) {
    if (scal[SC_DONE] != 0.f) return;
    int e = blockIdx.x * blockDim.x + threadIdx.x;
    if (e < 65536) {
        float s = 0.f;
        for (int kc = 0; kc < 8; ++kc) s += gradp[(size_t)kc * 65536 + e];
        grad[e] = s;
    }
}

// ---------------------------------------------------------------------------
// scalar momentum update: a_new = (1+sqrt(1+4a^2))/2 ; ratio = (a-1)/a_new
__global__ void k_scal(float* __restrict__ scal) {
    if (scal[SC_DONE] != 0.f) return;
    float a  = scal[SC_A];
    float an = (1.f + sqrtf(1.f + 4.f * a * a)) * 0.5f;
    scal[SC_RAT] = (a - 1.f) / an;
    scal[SC_A]   = an;
}

// A_new = soft_threshold(B - 2tau*(0.1*DtD@A + grad), tau*lam2)
// B_new = A + ratio*(A_new - A).  Block (32 l-threads, 8 cols).
__global__ void k_update(float* __restrict__ A, float* __restrict__ B,
                         const float* __restrict__ grad, const float* __restrict__ DtD,
                         const float* __restrict__ scal) {
    if (scal[SC_DONE] != 0.f) return;
    __shared__ float at[25][8];
    int l = threadIdx.x, c = threadIdx.y;
    int col = blockIdx.x * 8 + c;
    if (l < 25) at[l][c] = A[(size_t)l * D_FEAT + col];
    __syncthreads();
    if (l >= 25) return;
    float tau = scal[SC_TAU], lam2 = scal[SC_LAM2], ratio = scal[SC_RAT];
    float s = 0.f;
    for (int k = 0; k < 25; ++k) s += DtD[l * 25 + k] * at[k][c];
    float aold = at[l][c];
    float z = B[(size_t)l * D_FEAT + col] - 2.f * tau * (0.1f * s + grad[(size_t)l * D_FEAT + col]);
    float thr = tau * lam2;
    float an  = (z > thr) ? (z - thr) : ((z < -thr) ? (z + thr) : 0.f);
    A[(size_t)l * D_FEAT + col] = an;
    B[(size_t)l * D_FEAT + col] = aold + ratio * (an - aold);
}

// ---------------------------------------------------------------------------
// hankel tail (last 24 anti-diagonal means of AX) + NRMSE convergence check.
__global__ void k_tail(const float* __restrict__ AX, float* __restrict__ ytail,
                       float* __restrict__ scal) {
    if (scal[SC_DONE] != 0.f) return;
    __shared__ float axs[24];
    int t = threadIdx.x;   // 32 threads
    if (t < 24) {
        int s = M_SIZE + t;                 // anti-diagonal index
        float acc = 0.f;
        for (int l = t + 1; l <= 24; ++l) acc += AX[(size_t)l * M_SIZE + (s - l)];
        axs[t] = acc / (float)(24 - t);
    }
    __syncthreads();
    if (t == 0) {
        float sd = 0.f, sy = 0.f;
        for (int i = 0; i < 24; ++i) {
            float yt = ytail[i], d = yt - axs[i];
            sd += d * d; sy += yt * yt;
        }
        float conv = sqrtf(sd) / (sqrtf(sy) + 1e-12f);
        for (int i = 0; i < 24; ++i) ytail[i] = axs[i];
        if (conv <= 1e-5f) scal[SC_DONE] = 1.f;
    }
}

// ---------------------------------------------------------------------------
// output: [A@x (25x16384), A (25x2048)] flat (pad rows excluded -> contiguous)
__global__ void k_out(const float* __restrict__ AX, const float* __restrict__ A,
                      float* __restrict__ out) {
    int idx = blockIdx.x * blockDim.x + threadIdx.x;
    if (idx < 25 * M_SIZE)            out[idx] = AX[idx];
    else if (idx < 25 * M_SIZE + 25 * D_FEAT) out[idx] = A[idx - 25 * M_SIZE];
}

// ---------------------------------------------------------------------------
extern "C" void kernel_launch(void* const* d_in, const int* in_sizes, int n_in,
                              void* d_out, int out_size, void* d_ws, size_t ws_size,
                              hipStream_t stream) {
    (void)in_sizes; (void)n_in; (void)out_size; (void)ws_size;
    const float* x    = (const float*)d_in[0];
    const float* py   = (const float*)d_in[1];
    const float* mask = (const float*)d_in[2];
    float* ws    = (float*)d_ws;
    float* A     = ws + OFF_A;
    float* Bm    = ws + OFF_B;
    float* grad  = ws + OFF_GRAD;
    float* R     = ws + OFF_R;
    float* AX    = ws + OFF_AX;
    float* gradp = ws + OFF_GRADP;
    float* v     = ws + OFF_V;
    float* w     = ws + OFF_W;
    float* u     = ws + OFF_U;
    float* wpart = ws + OFF_WPART;
    float* DtD   = ws + OFF_DTD;
    float* ytail = ws + OFF_YTAIL;
    float* scal  = ws + OFF_SCAL;

    // init: A=B=0 (contiguous 2*65536), v = 1/sqrt(16384), Delta/DtD/eig/scalars
    k_fill<<<512, 256, 0, stream>>>(A, 2 * 65536, 0.f);
    k_fill<<<64, 256, 0, stream>>>(v, M_SIZE, 1.f / 128.f);
    k_init<<<1, 256, 0, stream>>>(ws);

    // power iteration on x^T x (30 rounds), then sigma^2 -> tau, lam2
    for (int it = 0; it < 30; ++it) {
        k_xv<<<D_FEAT, 256, 0, stream>>>(x, v, u);
        k_xtw<<<64, 256, 0, stream>>>(x, u, w, wpart);
        k_vnorm<<<64, 256, 0, stream>>>(w, wpart, v);
    }
    k_xv<<<D_FEAT, 256, 0, stream>>>(x, v, u);
    k_tau<<<1, 256, 0, stream>>>(u, scal);

    // 60 FISTA iterations; every kernel early-outs once converged (frozen state)
    for (int it = 0; it < 60; ++it) {
        k_gemm_ax<<<256, 128, 0, stream>>>(x, A, py, mask, R, scal, 0);   // resid
        k_gemm_rxt<<<dim3(32, 8), 128, 0, stream>>>(x, R, gradp, scal);   // grad (split-K)
        k_gsum<<<256, 256, 0, stream>>>(gradp, grad, scal);
        k_scal<<<1, 1, 0, stream>>>(scal);
        k_update<<<256, dim3(32, 8), 0, stream>>>(A, Bm, grad, DtD, scal);
        k_gemm_ax<<<256, 128, 0, stream>>>(x, A, py, mask, AX, scal, 1);  // A_new @ x
        k_tail<<<1, 32, 0, stream>>>(AX, ytail, scal);
    }
    k_out<<<1800, 256, 0, stream>>>(AX, A, (float*)d_out);
}